// MediumForceNet_37082747634271
// MI455X (gfx1250) — compile-verified
//
#include <hip/hip_runtime.h>

// ---------------------------------------------------------------------------
// MediumForceNet for MI455X (gfx1250, wave32).
// Pipeline: input proj (WMMA f32 GEMM) -> 3x GCN (WMMA GEMM + float4 atomic
// edge scatter + fused finalize) -> v/o projections (WMMA GEMM) -> head.
// q/k/softmax are mathematically identity for seq_len==1 and are elided.
// ---------------------------------------------------------------------------

#define N_NODES 50000
#define N_EDGES 800000
#define HID     128
#define KMAX    160   // layer-1 input dim (128 node + 32 coord feats)

typedef float v2f __attribute__((ext_vector_type(2)));
typedef float v8f __attribute__((ext_vector_type(8)));

// ---------------------------------------------------------------------------
// C[rows x 128] = A[rows x K] @ W[K x 128] (+ bias), fp32 via
// v_wmma_f32_16x16x4_f32.  One block = 256 threads = 8 wave32s.
// Block computes 16 rows x 128 cols; each wave owns one 16x16 tile.
// A is row-major [rows,K] (lda == K), so the 16-row slab is contiguous and
// staged in LDS with a straight cooperative copy.  W is [K,128] row-major,
// read through L2/WGP$ (it's tiny and shared by all blocks).
//
// Fragment layouts per CDNA5 ISA 7.12.2:
//   A 16x4 (2 VGPR): lanes 0-15 -> M=lane, K=k+0/k+1 ; lanes 16-31 -> k+2/k+3
//   B 4x16 (2 VGPR): mirrored (row K striped across lanes within a VGPR)
//   C/D 16x16 (8 VGPR): vgpr r -> M = r (+8 for lanes 16-31), N = lane&15
// ---------------------------------------------------------------------------
__global__ __launch_bounds__(256) void gemm16_wmma_f32(
    const float* __restrict__ A, const float* __restrict__ W,
    const float* __restrict__ bias, float* __restrict__ C,
    int K, int ldc, int colOffset)
{
    __shared__ float As[16 * KMAX];

    const int row0 = blockIdx.x * 16;
    const int tid  = threadIdx.x;

    // contiguous cooperative copy of 16 rows of A into LDS
    const float* Ablk = A + (size_t)row0 * (size_t)K;
    const int    tot  = 16 * K;
    for (int idx = tid; idx < tot; idx += 256) As[idx] = Ablk[idx];
    __syncthreads();

    const int lane = tid & 31;
    const int wave = tid >> 5;
    const int col0 = wave << 4;          // 8 waves cover cols 0..127
    const int n    = lane & 15;
    const int m    = lane & 15;
    const int koff = (lane < 16) ? 0 : 2;

    v8f c = {};                          // zero accumulator
    for (int k = 0; k < K; k += 4) {
        v2f a, b;
        const int kk = k + koff;
        a.x = As[m * K + kk];            // ds_load (pairs merge to b64)
        a.y = As[m * K + kk + 1];
        b.x = W[(size_t)kk       * HID + col0 + n];
        b.y = W[(size_t)(kk + 1) * HID + col0 + n];
        c = __builtin_amdgcn_wmma_f32_16x16x4_f32(
                /*neg_a=*/false, a, /*neg_b=*/false, b,
                /*c_mod=*/(short)0, c, /*reuse_a=*/false, /*reuse_b=*/false);
    }

    const float badd  = bias ? bias[col0 + n] : 0.0f;
    const int   rbase = row0 + ((lane >> 4) << 3);   // +8 for upper half-wave
#pragma unroll
    for (int r = 0; r < 8; ++r)
        C[(size_t)(rbase + r) * ldc + colOffset + col0 + n] = c[r] + badd;
}

// h0[:,128:160] = coords @ coord_w + coord_b   (coord_w is [2,32] row-major)
__global__ void coord_proj_kernel(const float* __restrict__ coords,
                                  const float* __restrict__ cw,
                                  const float* __restrict__ cb,
                                  float* __restrict__ h0)
{
    int i = blockIdx.x * blockDim.x + threadIdx.x;
    if (i >= N_NODES * 32) return;
    int node = i >> 5, j = i & 31;
    float x0 = coords[node * 2 + 0];
    float x1 = coords[node * 2 + 1];
    h0[(size_t)node * KMAX + HID + j] = x0 * cw[j] + x1 * cw[32 + j] + cb[j];
}

__global__ void deg_init_kernel(float* __restrict__ deg)
{
    int i = blockIdx.x * blockDim.x + threadIdx.x;
    if (i < N_NODES) deg[i] = 1.0f;      // self loop
}

__global__ void deg_count_kernel(const int* __restrict__ dst,
                                 float* __restrict__ deg)
{
    int e = blockIdx.x * blockDim.x + threadIdx.x;
    if (e < N_EDGES) atomicAdd(&deg[dst[e]], 1.0f);
}

__global__ void deg_rsqrt_kernel(const float* __restrict__ deg,
                                 float* __restrict__ dinv)
{
    int i = blockIdx.x * blockDim.x + threadIdx.x;
    if (i < N_NODES) dinv[i] = rsqrtf(deg[i]);
}

__global__ void zero_nodes_kernel(float4* __restrict__ agg)
{
    int i = blockIdx.x * blockDim.x + threadIdx.x;
    if (i < N_NODES * 32) agg[i] = make_float4(0.f, 0.f, 0.f, 0.f);
}

// agg[dst] += hw[src] * dinv[src]*dinv[dst]; 32 threads/edge, float4 each.
__global__ void edge_scatter_kernel(const float* __restrict__ hw,
                                    const int* __restrict__ src,
                                    const int* __restrict__ dst,
                                    const float* __restrict__ dinv,
                                    float* __restrict__ agg)
{
    long long gid = (long long)blockIdx.x * blockDim.x + threadIdx.x;
    long long e   = gid >> 5;
    if (e >= N_EDGES) return;
    const int f = (int)(gid & 31) << 2;          // feature quad
    const int s = src[e];
    const int d = dst[e];
    const float nrm = dinv[s] * dinv[d];
    const float4 msg = *(const float4*)&hw[(size_t)s * HID + f];
    float* out = &agg[(size_t)d * HID + f];
    atomicAdd(out + 0, msg.x * nrm);
    atomicAdd(out + 1, msg.y * nrm);
    atomicAdd(out + 2, msg.z * nrm);
    atomicAdd(out + 3, msg.w * nrm);
}

// h = relu(agg + hw * dinv^2 + b)
__global__ void gcn_finalize_kernel(const float* __restrict__ agg,
                                    const float* __restrict__ hw,
                                    const float* __restrict__ dinv,
                                    const float* __restrict__ b,
                                    float* __restrict__ h)
{
    long long idx = (long long)blockIdx.x * blockDim.x + threadIdx.x;
    if (idx >= (long long)N_NODES * HID) return;
    const int i = (int)(idx >> 7);
    const int f = (int)(idx & 127);
    const float di = dinv[i];
    const float v = agg[idx] + hw[idx] * (di * di) + b[f];
    h[idx] = fmaxf(v, 0.0f);
}

// out[i,:] = o[i,:] @ head_w[128,2] + head_b
__global__ void head_kernel(const float* __restrict__ o,
                            const float* __restrict__ hw2,
                            const float* __restrict__ hb2,
                            float* __restrict__ out)
{
    int i = blockIdx.x * blockDim.x + threadIdx.x;
    if (i >= N_NODES) return;
    float a0 = hb2[0], a1 = hb2[1];
    const float* row = o + (size_t)i * HID;
#pragma unroll 4
    for (int f = 0; f < HID; ++f) {
        const float t = row[f];
        a0 += t * hw2[f * 2 + 0];
        a1 += t * hw2[f * 2 + 1];
    }
    out[i * 2 + 0] = a0;
    out[i * 2 + 1] = a1;
}

// ---------------------------------------------------------------------------
extern "C" void kernel_launch(void* const* d_in, const int* in_sizes, int n_in,
                              void* d_out, int out_size, void* d_ws, size_t ws_size,
                              hipStream_t stream)
{
    const float* x       = (const float*)d_in[0];
    const float* coords  = (const float*)d_in[1];
    const int*   eidx    = (const int*)  d_in[2];
    const float* node_w  = (const float*)d_in[3];
    const float* node_b  = (const float*)d_in[4];
    const float* coord_w = (const float*)d_in[5];
    const float* coord_b = (const float*)d_in[6];
    const float* g1_w    = (const float*)d_in[7];
    const float* g1_b    = (const float*)d_in[8];
    const float* g2_w    = (const float*)d_in[9];
    const float* g2_b    = (const float*)d_in[10];
    const float* g3_w    = (const float*)d_in[11];
    const float* g3_b    = (const float*)d_in[12];
    // 13..16: wq,bq,wk,bk -- mathematically dead (softmax over one key == 1)
    const float* wv      = (const float*)d_in[17];
    const float* bv      = (const float*)d_in[18];
    const float* wo      = (const float*)d_in[19];
    const float* bo      = (const float*)d_in[20];
    const float* head_w  = (const float*)d_in[21];
    const float* head_b  = (const float*)d_in[22];

    const int* src = eidx;             // edge_index[0]
    const int* dst = eidx + N_EDGES;   // edge_index[1]

    // workspace carve-up (floats)
    float* ws   = (float*)d_ws;
    float* h0   = ws;                              // N*160
    float* deg  = h0  + (size_t)N_NODES * KMAX;    // N
    float* dinv = deg + N_NODES;                   // N
    float* hw   = dinv + N_NODES;                  // N*128
    float* agg  = hw  + (size_t)N_NODES * HID;     // N*128
    float* hA   = agg + (size_t)N_NODES * HID;     // N*128
    float* hB   = hA  + (size_t)N_NODES * HID;     // N*128
    float* out  = (float*)d_out;                   // N*2

    const int gemmGrid = N_NODES / 16;             // 50000/16 == 3125 exact
    const int nThreads = 256;
    const int gN    = (N_NODES + nThreads - 1) / nThreads;
    const int gE    = (N_EDGES + nThreads - 1) / nThreads;
    const int gN32  = (N_NODES * 32 + nThreads - 1) / nThreads;
    const int gN128 = (int)(((long long)N_NODES * HID + nThreads - 1) / nThreads);
    const int gE32  = (int)(((long long)N_EDGES * 32 + nThreads - 1) / nThreads);

    // degrees (independent of features)
    deg_init_kernel <<<gN, nThreads, 0, stream>>>(deg);
    deg_count_kernel<<<gE, nThreads, 0, stream>>>(dst, deg);
    deg_rsqrt_kernel<<<gN, nThreads, 0, stream>>>(deg, dinv);

    // input projection -> h0 [N,160]
    gemm16_wmma_f32<<<gemmGrid, nThreads, 0, stream>>>(x, node_w, node_b, h0,
                                                       /*K=*/128, /*ldc=*/KMAX, /*colOff=*/0);
    coord_proj_kernel<<<gN32, nThreads, 0, stream>>>(coords, coord_w, coord_b, h0);

    // ---- GCN layer 1: h0[N,160] -> hA[N,128]
    gemm16_wmma_f32<<<gemmGrid, nThreads, 0, stream>>>(h0, g1_w, nullptr, hw, 160, HID, 0);
    zero_nodes_kernel <<<gN32,  nThreads, 0, stream>>>((float4*)agg);
    edge_scatter_kernel<<<gE32, nThreads, 0, stream>>>(hw, src, dst, dinv, agg);
    gcn_finalize_kernel<<<gN128,nThreads, 0, stream>>>(agg, hw, dinv, g1_b, hA);

    // ---- GCN layer 2: hA -> hB
    gemm16_wmma_f32<<<gemmGrid, nThreads, 0, stream>>>(hA, g2_w, nullptr, hw, 128, HID, 0);
    zero_nodes_kernel <<<gN32,  nThreads, 0, stream>>>((float4*)agg);
    edge_scatter_kernel<<<gE32, nThreads, 0, stream>>>(hw, src, dst, dinv, agg);
    gcn_finalize_kernel<<<gN128,nThreads, 0, stream>>>(agg, hw, dinv, g2_b, hB);

    // ---- GCN layer 3: hB -> hA
    gemm16_wmma_f32<<<gemmGrid, nThreads, 0, stream>>>(hB, g3_w, nullptr, hw, 128, HID, 0);
    zero_nodes_kernel <<<gN32,  nThreads, 0, stream>>>((float4*)agg);
    edge_scatter_kernel<<<gE32, nThreads, 0, stream>>>(hw, src, dst, dinv, agg);
    gcn_finalize_kernel<<<gN128,nThreads, 0, stream>>>(agg, hw, dinv, g3_b, hA);

    // ---- attention collapses to out_proj(v_proj(h)) for seq_len == 1
    gemm16_wmma_f32<<<gemmGrid, nThreads, 0, stream>>>(hA, wv, bv, hB, 128, HID, 0); // v
    gemm16_wmma_f32<<<gemmGrid, nThreads, 0, stream>>>(hB, wo, bo, hw, 128, HID, 0); // o

    // ---- regression head [128 -> 2]
    head_kernel<<<gN, nThreads, 0, stream>>>(hw, head_w, head_b, out);
}